// GCN_86990267613312
// MI455X (gfx1250) — compile-verified
//
#include <hip/hip_runtime.h>
#include <hip/hip_bf16.h>

#define DFEAT 128

typedef __attribute__((ext_vector_type(2))) float v2f;
typedef __attribute__((ext_vector_type(8))) float v8f;
typedef __attribute__((ext_vector_type(4))) int   v4i;

// async global->LDS builtin takes (v4i AS1*, v4i AS3*, imm offset, imm cpol)
typedef __attribute__((address_space(1))) v4i gv4i;
typedef __attribute__((address_space(3))) v4i lv4i;

#if defined(__AMDGCN__) && __has_builtin(__builtin_amdgcn_global_load_async_to_lds_b128)
#define HAVE_ASYNC_LDS 1
#else
#define HAVE_ASYNC_LDS 0
#endif

// ---------------------------------------------------------------------------
// H = X @ W   (X: [nrows x 128], W: [128 x 128], H: [nrows x 128])
//
// Block = 256 threads (8 waves).  The block stages its 16x128 A-tile (8 KB)
// and the full 128x128 W (64 KB) into LDS via GLOBAL_LOAD_ASYNC_TO_LDS_B128
// (tracked by ASYNCcnt), then each wave computes one 16x16 output tile with
// 32 chained V_WMMA_F32_16X16X4_F32, operands sourced from LDS.
// 72 KB LDS/block -> 4 blocks per 320 KB WGP.
// ---------------------------------------------------------------------------
__global__ void gcn_gemm_wmma(const float* __restrict__ X,
                              const float* __restrict__ W,
                              float* __restrict__ H) {
  __shared__ float Xs[16 * DFEAT];          // 8 KB  A tile, shared by 8 waves
  __shared__ float Ws[DFEAT * DFEAT];       // 64 KB full weight matrix

  const int t    = threadIdx.x;             // 0..255
  const int lane = t & 31;
  const int wave = t >> 5;                  // 0..7 -> 16-column tile
  const int row0 = blockIdx.x * 16;         // 100000 % 16 == 0, no guard
  const int col0 = wave * 16;

  const float4* __restrict__ xg = (const float4*)(X + (size_t)row0 * DFEAT);
  const float4* __restrict__ wg = (const float4*)W;

#if HAVE_ASYNC_LDS
  {
    float4* xs = (float4*)Xs;
    float4* ws = (float4*)Ws;
    // Casts between address spaces must go through an integer in HIP clang.
    // Generic shared pointer low 32 bits == LDS offset (ISA aperture rule).
    // X tile: 512 float4 -> 2 per thread
    for (int i = t; i < (16 * DFEAT) / 4; i += 256)
      __builtin_amdgcn_global_load_async_to_lds_b128(
          (gv4i*)(unsigned long long)(xg + i),
          (lv4i*)(unsigned int)(unsigned long long)(xs + i), 0, 0);
    // W: 4096 float4 -> 16 per thread
    for (int i = t; i < (DFEAT * DFEAT) / 4; i += 256)
      __builtin_amdgcn_global_load_async_to_lds_b128(
          (gv4i*)(unsigned long long)(wg + i),
          (lv4i*)(unsigned int)(unsigned long long)(ws + i), 0, 0);
    asm volatile("s_wait_asynccnt 0x0" ::: "memory");
  }
#else
  for (int i = t; i < (16 * DFEAT) / 4; i += 256) ((float4*)Xs)[i] = xg[i];
  for (int i = t; i < (DFEAT * DFEAT) / 4; i += 256) ((float4*)Ws)[i] = wg[i];
#endif
  __syncthreads();

  // WMMA operand layout (f32 16x16x4):
  //  A: lanes 0-15 row=lane K={0,1}; lanes 16-31 row=lane-16 K={2,3}
  //  B: mirrored per column; C/D: vgpr i -> M=i / i+8 per lane half.
  const int r  = lane & 15;
  const int kh = (lane >> 4) * 2;           // 0 or 2

  const float* __restrict__ xrow = Xs + r * DFEAT;
  const float* __restrict__ wcol = Ws + col0 + r;

  v8f acc = {};
#pragma unroll
  for (int k = 0; k < DFEAT; k += 4) {
    v2f a, b;
    a.x = xrow[k + kh];                     // ds_load_b64 pair
    a.y = xrow[k + kh + 1];
    b.x = wcol[(size_t)(k + kh) * DFEAT];   // ds_load_b32 x2
    b.y = wcol[(size_t)(k + kh + 1) * DFEAT];
    acc = __builtin_amdgcn_wmma_f32_16x16x4_f32(
        false, a, false, b, (short)0, acc, false, false);
  }

  const int m_off = (lane >> 4) * 8;
  float* __restrict__ out = H + (size_t)row0 * DFEAT + col0 + r;
#pragma unroll
  for (int i = 0; i < 8; ++i)
    out[(size_t)(m_off + i) * DFEAT] = acc[i];
}

// ---------------------------------------------------------------------------
// Degree / normalization
// ---------------------------------------------------------------------------
__global__ void init_deg(float* __restrict__ deg, int n) {
  int i = blockIdx.x * blockDim.x + threadIdx.x;
  if (i < n) deg[i] = 1.0f;                 // self-loop contribution
}

__global__ void count_deg(const long long* __restrict__ dst,
                          float* __restrict__ deg, long long nedges) {
  long long e = (long long)blockIdx.x * blockDim.x + threadIdx.x;
  if (e < nedges) atomicAdd(&deg[dst[e]], 1.0f);
}

__global__ void compute_dinv(const float* __restrict__ deg,
                             float* __restrict__ dinv, int n) {
  int i = blockIdx.x * blockDim.x + threadIdx.x;
  if (i < n) dinv[i] = rsqrtf(deg[i]);      // deg >= 1 always (self-loop)
}

// ---------------------------------------------------------------------------
// agg[i,:] = h[i,:] * dinv[i]^2   (self-loop edge, also zero-initializes agg)
// ---------------------------------------------------------------------------
__global__ void init_agg_selfloop(const float* __restrict__ h,
                                  const float* __restrict__ dinv,
                                  float* __restrict__ agg, long long total) {
  long long idx = (long long)blockIdx.x * blockDim.x + threadIdx.x;
  if (idx < total) {
    float di = dinv[idx >> 7];              // node = idx / 128
    agg[idx] = h[idx] * di * di;
  }
}

// ---------------------------------------------------------------------------
// Edge aggregation: one wave32 per edge; lane l owns float4 [4l..4l+3].
// agg[dst,:] += h[src,:] * dinv[src]*dinv[dst]  via global_atomic_add_f32.
// Bandwidth-bound hot loop (~2 GB/layer of edge traffic @ 23.3 TB/s).
// ---------------------------------------------------------------------------
__global__ void edge_scatter(const long long* __restrict__ src,
                             const long long* __restrict__ dst,
                             const float* __restrict__ dinv,
                             const float* __restrict__ h,
                             float* __restrict__ agg, long long nedges) {
  long long e = (long long)blockIdx.x * (blockDim.x >> 5) + (threadIdx.x >> 5);
  if (e >= nedges) return;
  const int lane = threadIdx.x & 31;

  const long long s = src[e];
  const long long d = dst[e];
  const float w = dinv[s] * dinv[d];

  const float4* __restrict__ hp = (const float4*)(h + (size_t)s * DFEAT);
  float4 v = hp[lane];                      // global_load_b128 gather

  float* __restrict__ ap = agg + (size_t)d * DFEAT + lane * 4;
  atomicAdd(ap + 0, v.x * w);
  atomicAdd(ap + 1, v.y * w);
  atomicAdd(ap + 2, v.z * w);
  atomicAdd(ap + 3, v.w * w);
}

// ---------------------------------------------------------------------------
// out = relu(agg + b)
// ---------------------------------------------------------------------------
__global__ void relu_bias(const float* __restrict__ agg,
                          const float* __restrict__ b,
                          float* __restrict__ out, long long total) {
  long long idx = (long long)blockIdx.x * blockDim.x + threadIdx.x;
  if (idx < total)
    out[idx] = fmaxf(agg[idx] + b[idx & (DFEAT - 1)], 0.0f);
}

// ---------------------------------------------------------------------------
// Orchestration
// ---------------------------------------------------------------------------
extern "C" void kernel_launch(void* const* d_in, const int* in_sizes, int n_in,
                              void* d_out, int out_size, void* d_ws, size_t ws_size,
                              hipStream_t stream) {
  const float*      x0 = (const float*)d_in[0];          // [N, 128]
  const long long*  ei = (const long long*)d_in[1];      // [2, E] int64
  const float*      W1 = (const float*)d_in[2];
  const float*      b1 = (const float*)d_in[3];
  const float*      W2 = (const float*)d_in[4];
  const float*      b2 = (const float*)d_in[5];
  const float*      W3 = (const float*)d_in[6];
  const float*      b3 = (const float*)d_in[7];

  const int       n      = in_sizes[0] / DFEAT;          // 100000
  const long long nedges = (long long)in_sizes[1] / 2;   // 1,600,000
  const long long total  = (long long)n * DFEAT;

  const long long* src = ei;
  const long long* dst = ei + nedges;

  float* deg  = (float*)d_ws;
  float* dinv = deg  + n;
  float* h    = dinv + n;                                // [N,128]
  float* agg  = h    + total;                            // [N,128]
  float* xbuf = agg  + total;                            // [N,128]
  float* yout = (float*)d_out;

  const int T = 256;
  const int nodeBlk  = (n + T - 1) / T;
  const int elemBlk  = (int)((total + T - 1) / T);
  const int edgeBlkT = (int)((nedges + T - 1) / T);      // 1 thread / edge
  const int edgeBlkW = (int)((nedges + 7) / 8);          // 1 wave / edge
  const int gemmBlk  = n / 16;                           // 6250 row tiles

  // --- normalization (shared by all 3 layers) ---
  init_deg<<<nodeBlk, T, 0, stream>>>(deg, n);
  count_deg<<<edgeBlkT, T, 0, stream>>>(dst, deg, nedges);
  compute_dinv<<<nodeBlk, T, 0, stream>>>(deg, dinv, n);

  // --- layer 1 ---
  gcn_gemm_wmma<<<gemmBlk, T, 0, stream>>>(x0, W1, h);
  init_agg_selfloop<<<elemBlk, T, 0, stream>>>(h, dinv, agg, total);
  edge_scatter<<<edgeBlkW, T, 0, stream>>>(src, dst, dinv, h, agg, nedges);
  relu_bias<<<elemBlk, T, 0, stream>>>(agg, b1, xbuf, total);

  // --- layer 2 ---
  gcn_gemm_wmma<<<gemmBlk, T, 0, stream>>>(xbuf, W2, h);
  init_agg_selfloop<<<elemBlk, T, 0, stream>>>(h, dinv, agg, total);
  edge_scatter<<<edgeBlkW, T, 0, stream>>>(src, dst, dinv, h, agg, nedges);
  relu_bias<<<elemBlk, T, 0, stream>>>(agg, b2, xbuf, total);

  // --- layer 3 ---
  gcn_gemm_wmma<<<gemmBlk, T, 0, stream>>>(xbuf, W3, h);
  init_agg_selfloop<<<elemBlk, T, 0, stream>>>(h, dinv, agg, total);
  edge_scatter<<<edgeBlkW, T, 0, stream>>>(src, dst, dinv, h, agg, nedges);
  relu_bias<<<elemBlk, T, 0, stream>>>(agg, b3, yout, total);
}